// InteractionPPBlock_3822520894068
// MI455X (gfx1250) — compile-verified
//
#include <hip/hip_runtime.h>
#include <hip/hip_bf16.h>

typedef float v2f __attribute__((ext_vector_type(2)));
typedef float v8f __attribute__((ext_vector_type(8)));

#define HDIM 128
#define LSTRIDE 132   // 128 + 4 pad: row-to-row LDS bank shift of 4 -> conflict-free

static __device__ __forceinline__ v8f wmma4(v2f a, v2f b, v8f c) {
  // D = A(16x4 f32) * B(4x16 f32) + C(16x16 f32), fp32-exact path
  return __builtin_amdgcn_wmma_f32_16x16x4_f32(false, a, false, b, (short)0, c, false, false);
}

static __device__ __forceinline__ float silu(float x) {
  return x / (1.0f + __expf(-x));
}

// ---------------- out = x -------------------------------------------------
__global__ __launch_bounds__(256) void k_copy(const float* __restrict__ x,
                                              float* __restrict__ out, int n4) {
  int i = blockIdx.x * 256 + threadIdx.x;
  if (i < n4) ((float4*)out)[i] = ((const float4*)x)[i];
}

// ---------------- Wc = W_t1 @ W_down ; bc = W_t1 @ b_down ; extra = 0 -----
__global__ __launch_bounds__(256) void k_combine(const float* __restrict__ Wt1,
                                                 const float* __restrict__ Wdown,
                                                 const float* __restrict__ bdown,
                                                 float* __restrict__ Wc,
                                                 float* __restrict__ bc,
                                                 float* __restrict__ extra) {
  int id = blockIdx.x * 256 + threadIdx.x;          // 0..16383
  int j = id >> 7, k = id & 127;
  float s = 0.0f;
  for (int m = 0; m < HDIM; ++m) s += Wt1[j * HDIM + m] * Wdown[m * HDIM + k];
  Wc[j * HDIM + k] = s;
  if (id < HDIM) {
    float b = 0.0f;
    for (int kk = 0; kk < HDIM; ++kk) b += Wt1[id * HDIM + kk] * bdown[kk];
    bc[id] = b;
    extra[id] = 0.0f;
  }
}

// ---------------- node kernel: x_up and t, weights resident in LDS --------
__global__ __launch_bounds__(256) void k_node(const float* __restrict__ x,
                                              const float* __restrict__ Wup,
                                              const float* __restrict__ bup,
                                              const float* __restrict__ Wt2,
                                              const float* __restrict__ bt2,
                                              const float* __restrict__ Wc,
                                              const float* __restrict__ bc,
                                              float* __restrict__ xup,
                                              float* __restrict__ t_out,
                                              int nTiles) {
  __shared__ float lWup[HDIM * LSTRIDE];
  __shared__ float lWc [HDIM * LSTRIDE];
  __shared__ float lWt2[HDIM * LSTRIDE];
  __shared__ float xs[16 * LSTRIDE];
  __shared__ float ss[16 * LSTRIDE];

  const int tid = threadIdx.x;
  for (int i = tid; i < HDIM * HDIM; i += 256) {
    int r = i >> 7, c = i & 127;
    lWup[r * LSTRIDE + c] = Wup[i];
    lWc [r * LSTRIDE + c] = Wc[i];
    lWt2[r * LSTRIDE + c] = Wt2[i];
  }

  const int lane = tid & 31, wave = tid >> 5;
  const int hlf = lane >> 4, l16 = lane & 15;
  const int n = wave * 16 + l16;          // output column for this lane
  const int koff = hlf * 2;               // K sub-offset per half-wave
  const float bupN = bup[n], bt2N = bt2[n], bcN = bc[n];
  __syncthreads();

  for (int tile = blockIdx.x; tile < nTiles; tile += gridDim.x) {
    __syncthreads();
    const int r0 = tile << 4;
    for (int i = tid; i < 16 * HDIM; i += 256) {
      int r = i >> 7, c = i & 127;
      xs[r * LSTRIDE + c] = x[(r0 + r) * HDIM + c];
    }
    __syncthreads();

    v8f c1 = {0.f,0.f,0.f,0.f,0.f,0.f,0.f,0.f};
    v8f c2 = {0.f,0.f,0.f,0.f,0.f,0.f,0.f,0.f};
    for (int k0 = 0; k0 < HDIM; k0 += 4) {
      v2f a, b1, b2;
      a.x  = xs[l16 * LSTRIDE + k0 + koff];
      a.y  = xs[l16 * LSTRIDE + k0 + koff + 1];
      b1.x = lWup[n * LSTRIDE + k0 + koff];
      b1.y = lWup[n * LSTRIDE + k0 + koff + 1];
      b2.x = lWc [n * LSTRIDE + k0 + koff];
      b2.y = lWc [n * LSTRIDE + k0 + koff + 1];
      c1 = wmma4(a, b1, c1);
      c2 = wmma4(a, b2, c2);
    }
#pragma unroll
    for (int v = 0; v < 8; ++v) {
      const int m = v + hlf * 8;
      xup[(r0 + m) * HDIM + n] = c1[v] + bupN;
      ss[m * LSTRIDE + n] = silu(c2[v] + bcN);
    }
    __syncthreads();

    v8f c3 = {0.f,0.f,0.f,0.f,0.f,0.f,0.f,0.f};
    for (int k0 = 0; k0 < HDIM; k0 += 4) {
      v2f a, b;
      a.x = ss[l16 * LSTRIDE + k0 + koff];
      a.y = ss[l16 * LSTRIDE + k0 + koff + 1];
      b.x = lWt2[n * LSTRIDE + k0 + koff];
      b.y = lWt2[n * LSTRIDE + k0 + koff + 1];
      c3 = wmma4(a, b, c3);
    }
#pragma unroll
    for (int v = 0; v < 8; ++v) {
      const int m = v + hlf * 8;
      t_out[(r0 + m) * HDIM + n] = c3[v] + bt2N;
    }
  }
}

// ---------------- triplet reduction: extra[h] = sum t[trip1]*sbf_emb ------
__global__ __launch_bounds__(256) void k_trip(const float* __restrict__ sbf,
                                              const int* __restrict__ trip,
                                              const float* __restrict__ Ws1,
                                              const float* __restrict__ Ws2,
                                              const float* __restrict__ t_ws,
                                              float* __restrict__ extra,
                                              int nTiles) {
  __shared__ float lWs1[8 * 42];
  __shared__ float sb[16 * 44];
  __shared__ float hs[16 * 12];
  __shared__ int tIx[16];

  const int tid = threadIdx.x;
  for (int i = tid; i < 8 * 42; i += 256) lWs1[i] = Ws1[i];

  const int lane = tid & 31, wave = tid >> 5;
  const int hlf = lane >> 4, l16 = lane & 15;
  const int n = wave * 16 + l16;
  const int koff = hlf * 2;

  v2f bA, bB;                               // W_sbf2 [128][8] slab, kept in regs
  bA.x = Ws2[n * 8 + koff];     bA.y = Ws2[n * 8 + koff + 1];
  bB.x = Ws2[n * 8 + 4 + koff]; bB.y = Ws2[n * 8 + 4 + koff + 1];

  v8f acc = {0.f,0.f,0.f,0.f,0.f,0.f,0.f,0.f};
  __syncthreads();

  for (int tile = blockIdx.x; tile < nTiles; tile += gridDim.x) {
    __syncthreads();
    const int e0 = tile << 4;
    const float* sp = sbf + (size_t)e0 * 42;
    const bool pf = (tile + gridDim.x) < nTiles;
    for (int i = tid; i < 16 * 42; i += 256) {
      sb[(i / 42) * 44 + (i % 42)] = sp[i];
      if (pf) __builtin_prefetch(sp + (size_t)gridDim.x * 16 * 42 + i, 0, 1);
    }
    if (tid < 16) tIx[tid] = trip[(e0 + tid) * 3 + 1];
    __syncthreads();
    if (tid < 128) {
      const int e = tid >> 3, j = tid & 7;
      float h = 0.0f;
      for (int k = 0; k < 42; ++k) h += sb[e * 44 + k] * lWs1[j * 42 + k];
      hs[e * 12 + j] = silu(h);
    }
    __syncthreads();

    v2f a;
    v8f c = {0.f,0.f,0.f,0.f,0.f,0.f,0.f,0.f};
    a.x = hs[l16 * 12 + koff];     a.y = hs[l16 * 12 + koff + 1];
    c = wmma4(a, bA, c);
    a.x = hs[l16 * 12 + 4 + koff]; a.y = hs[l16 * 12 + 4 + koff + 1];
    c = wmma4(a, bB, c);
#pragma unroll
    for (int v = 0; v < 8; ++v) {
      const int m = v + hlf * 8;
      acc[v] += t_ws[(size_t)tIx[m] * HDIM + n] * c[v];
    }
  }
  float s = acc[0] + acc[1] + acc[2] + acc[3] + acc[4] + acc[5] + acc[6] + acc[7];
  s += __shfl_xor(s, 16);                   // lanes l and l+16 hold the same column n
  if (hlf == 0) atomicAdd(&extra[n], s);
}

// ---------------- edge kernel: rbf_emb * x_up[row] + extra -> scatter -----
__global__ __launch_bounds__(256) void k_edge(const float* __restrict__ rbf,
                                              const int* __restrict__ eidx,
                                              const float* __restrict__ Wr1,
                                              const float* __restrict__ Wr2,
                                              const float* __restrict__ xup,
                                              const float* __restrict__ extra,
                                              float* __restrict__ out,
                                              int nTiles, int E) {
  __shared__ float lWr1[48];
  __shared__ float hs[16 * 12];
  __shared__ int er[16];
  __shared__ int ec[16];

  const int tid = threadIdx.x;
  if (tid < 48) lWr1[tid] = Wr1[tid];

  const int lane = tid & 31, wave = tid >> 5;
  const int hlf = lane >> 4, l16 = lane & 15;
  const int n = wave * 16 + l16;
  const int koff = hlf * 2;

  v2f bA, bB;                               // W_rbf2 [128][8] slab in regs
  bA.x = Wr2[n * 8 + koff];     bA.y = Wr2[n * 8 + koff + 1];
  bB.x = Wr2[n * 8 + 4 + koff]; bB.y = Wr2[n * 8 + 4 + koff + 1];
  const float extraN = extra[n];
  __syncthreads();

  for (int tile = blockIdx.x; tile < nTiles; tile += gridDim.x) {
    __syncthreads();
    const int e0 = tile << 4;
    if (tid < 16)               er[tid] = eidx[e0 + tid];
    else if (tid < 32)          ec[tid - 16] = eidx[E + e0 + tid - 16];
    if (tid < 128) {
      const int e = tid >> 3, j = tid & 7;
      float h = 0.0f;
      for (int k = 0; k < 6; ++k) h += rbf[(size_t)(e0 + e) * 6 + k] * lWr1[j * 6 + k];
      hs[e * 12 + j] = silu(h);
    }
    __syncthreads();

    v2f a;
    v8f c = {0.f,0.f,0.f,0.f,0.f,0.f,0.f,0.f};
    a.x = hs[l16 * 12 + koff];     a.y = hs[l16 * 12 + koff + 1];
    c = wmma4(a, bA, c);
    a.x = hs[l16 * 12 + 4 + koff]; a.y = hs[l16 * 12 + 4 + koff + 1];
    c = wmma4(a, bB, c);
#pragma unroll
    for (int v = 0; v < 8; ++v) {
      const int m = v + hlf * 8;
      const int r = er[m], cc = ec[m];
      const float val = xup[(size_t)r * HDIM + n] * c[v] + extraN;
      atomicAdd(&out[(size_t)cc * HDIM + n], val);
    }
  }
}

extern "C" void kernel_launch(void* const* d_in, const int* in_sizes, int n_in,
                              void* d_out, int out_size, void* d_ws, size_t ws_size,
                              hipStream_t stream) {
  const float* x     = (const float*)d_in[0];
  const float* rbf   = (const float*)d_in[1];
  const float* sbf   = (const float*)d_in[2];
  const int*   eidx  = (const int*)d_in[3];
  const int*   trip  = (const int*)d_in[4];
  const float* Wr1   = (const float*)d_in[5];
  const float* Wr2   = (const float*)d_in[6];
  const float* Ws1   = (const float*)d_in[7];
  const float* Ws2   = (const float*)d_in[8];
  const float* Wt1   = (const float*)d_in[9];
  const float* Wt2   = (const float*)d_in[10];
  const float* bt2   = (const float*)d_in[11];
  const float* Wup   = (const float*)d_in[12];
  const float* bup   = (const float*)d_in[13];
  const float* Wdown = (const float*)d_in[14];
  const float* bdown = (const float*)d_in[15];
  float* out = (float*)d_out;

  const int N = in_sizes[0] / HDIM;
  const int E = in_sizes[1] / 6;
  const int T = in_sizes[2] / 42;

  float* ws    = (float*)d_ws;
  float* xup   = ws;
  float* t_ws  = xup + (size_t)N * HDIM;
  float* Wc    = t_ws + (size_t)N * HDIM;
  float* bc    = Wc + HDIM * HDIM;
  float* extra = bc + HDIM;

  // 1) out = x
  {
    int n4 = (N * HDIM) / 4;
    k_copy<<<(n4 + 255) / 256, 256, 0, stream>>>(x, out, n4);
  }
  // 2) combined weights + zero the triplet accumulator (every call)
  k_combine<<<(HDIM * HDIM) / 256, 256, 0, stream>>>(Wt1, Wdown, bdown, Wc, bc, extra);
  // 3) x_up and t for all nodes (WMMA, weights LDS-resident)
  k_node<<<1024, 256, 0, stream>>>(x, Wup, bup, Wt2, bt2, Wc, bc, xup, t_ws, N / 16);
  // 4) triplet reduction -> extra[128]
  k_trip<<<2048, 256, 0, stream>>>(sbf, trip, Ws1, Ws2, t_ws, extra, T / 16);
  // 5) edge embedding + scatter-add into out
  k_edge<<<4096, 256, 0, stream>>>(rbf, eidx, Wr1, Wr2, xup, extra, out, E / 16, E);
}